// GapAwareGNN_52063593562922
// MI455X (gfx1250) — compile-verified
//
#include <hip/hip_runtime.h>
#include <math.h>

// ---------------------------------------------------------------------------
// GapAwareGNN on MI455X (gfx1250).
// Dense node GEMMs use V_WMMA_F32_16X16X4_F32 (fp32 WMMA, wave32).
// Edge aggregation uses L2-resident atomics (f32 add, ordered-uint max).
// ---------------------------------------------------------------------------

typedef float v2f __attribute__((ext_vector_type(2)));
typedef float v8f __attribute__((ext_vector_type(8)));

__device__ __forceinline__ float leaky02(float x) { return x > 0.f ? x : 0.2f * x; }

// monotonic float <-> uint encoding so atomicMax(unsigned*) == float max
__device__ __forceinline__ unsigned enc_f(float f) {
  unsigned u = __float_as_uint(f);
  return (u >> 31) ? ~u : (u | 0x80000000u);
}
__device__ __forceinline__ float dec_f(unsigned e) {
  return (e >> 31) ? __uint_as_float(e & 0x7FFFFFFFu) : __uint_as_float(~e);
}

// ---------------------------------------------------------------------------
// Setup: per-branch scalar reductions of the rank-1 GAT parameters.
//   S_as[p][h] = sum_c W_p[h*16+c] * a_s[p][h][c]           (12 values)
//   S_ad[p][h] = sum_c W_p[h*16+c] * a_d[p][h][c]           (12 values)
//   T[p][j][h] = sum_c We_p[j*64 + h*16+c] * a_e[p][h][c]   (36 values)
// Const layout (floats): S_as at [p*4+h], S_ad at [16+p*4+h], T at [32+p*12+j*4+h]
// ---------------------------------------------------------------------------
__global__ void k_setup(float* c,
    const float* W0, const float* as0, const float* ad0, const float* ae0, const float* We0,
    const float* W1, const float* as1, const float* ad1, const float* ae1, const float* We1,
    const float* W2p, const float* as2, const float* ad2, const float* ae2, const float* We2) {
  const float* W[3]  = {W0, W1, W2p};
  const float* As[3] = {as0, as1, as2};
  const float* Ad[3] = {ad0, ad1, ad2};
  const float* Ae[3] = {ae0, ae1, ae2};
  const float* We[3] = {We0, We1, We2};
  int t = threadIdx.x;
  if (t < 12) {
    int p = t / 4, h = t % 4; float s = 0.f;
    for (int cc = 0; cc < 16; ++cc) s += W[p][h * 16 + cc] * As[p][h * 16 + cc];
    c[p * 4 + h] = s;
  } else if (t < 24) {
    int q = t - 12; int p = q / 4, h = q % 4; float s = 0.f;
    for (int cc = 0; cc < 16; ++cc) s += W[p][h * 16 + cc] * Ad[p][h * 16 + cc];
    c[16 + p * 4 + h] = s;
  } else if (t < 60) {
    int q = t - 24; int p = q / 12, j = (q % 12) / 4, h = q % 4; float s = 0.f;
    for (int cc = 0; cc < 16; ++cc) s += We[p][j * 64 + h * 16 + cc] * Ae[p][h * 16 + cc];
    c[32 + p * 12 + j * 4 + h] = s;
  }
}

// ---------------------------------------------------------------------------
// Branch stage 1: masked segment-mean numerators of edge_attr per dst node.
// ---------------------------------------------------------------------------
__global__ void k_attr_accum(const int* __restrict__ ei, const float* __restrict__ ea,
                             const unsigned char* m0, const unsigned char* m1,
                             const unsigned char* m2, float* attr_sum, float* cnt,
                             int N, int E) {
  int e = blockIdx.x * blockDim.x + threadIdx.x;
  if (e >= E) return;
  int dst = ei[E + e];
  float a0 = ea[e * 3 + 0], a1 = ea[e * 3 + 1], a2 = ea[e * 3 + 2];
  const unsigned char* M[3] = {m0, m1, m2};
  for (int p = 0; p < 3; ++p) {
    if (M[p][e]) {
      float* s = attr_sum + (size_t)p * N * 3 + (size_t)dst * 3;
      atomicAdd(&s[0], a0); atomicAdd(&s[1], a1); atomicAdd(&s[2], a2);
      atomicAdd(&cnt[(size_t)p * N + dst], 1.0f);
    }
  }
}

// ---------------------------------------------------------------------------
// Branch stage 2: per-node self-loop score, init segment-max; zero den/s.
// ---------------------------------------------------------------------------
__global__ void k_branch_node(const float* __restrict__ x, const float* __restrict__ c,
                              const float* __restrict__ attr_sum, const float* __restrict__ cnt,
                              unsigned* m_br, float* den, float* s_br, int N) {
  int i = blockIdx.x * blockDim.x + threadIdx.x;
  if (i >= N) return;
  float xi = x[i];
  for (int p = 0; p < 3; ++p) {
    float la[3];
    float ct = fmaxf(cnt[(size_t)p * N + i], 1.0f);
    for (int j = 0; j < 3; ++j) la[j] = attr_sum[(size_t)p * N * 3 + (size_t)i * 3 + j] / ct;
    for (int h = 0; h < 4; ++h) {
      float ls = xi * (c[p * 4 + h] + c[16 + p * 4 + h]);
      for (int j = 0; j < 3; ++j) ls += la[j] * c[32 + p * 12 + j * 4 + h];
      ls = leaky02(ls);
      size_t idx = (size_t)p * N * 4 + (size_t)i * 4 + h;
      m_br[idx] = enc_f(ls);
      den[idx] = 0.f;
      s_br[idx] = 0.f;
    }
  }
}

// ---------------------------------------------------------------------------
// Branch stage 3: per-edge attention score -> atomic segment max.
// ---------------------------------------------------------------------------
__global__ void k_branch_edge_max(const float* __restrict__ x, const int* __restrict__ ei,
                                  const float* __restrict__ ea,
                                  const unsigned char* m0, const unsigned char* m1,
                                  const unsigned char* m2, const float* __restrict__ c,
                                  unsigned* m_br, int N, int E) {
  int e = blockIdx.x * blockDim.x + threadIdx.x;
  if (e >= E) return;
  int src = ei[e], dst = ei[E + e];
  float xs = x[src], xd = x[dst];
  float a0 = ea[e * 3 + 0], a1 = ea[e * 3 + 1], a2 = ea[e * 3 + 2];
  const unsigned char* M[3] = {m0, m1, m2};
  for (int p = 0; p < 3; ++p) {
    if (!M[p][e]) continue;
    for (int h = 0; h < 4; ++h) {
      float al = xs * c[p * 4 + h] + xd * c[16 + p * 4 + h]
               + a0 * c[32 + p * 12 + 0 * 4 + h]
               + a1 * c[32 + p * 12 + 1 * 4 + h]
               + a2 * c[32 + p * 12 + 2 * 4 + h];
      al = leaky02(al);
      atomicMax(&m_br[(size_t)p * N * 4 + (size_t)dst * 4 + h], enc_f(al));
    }
  }
}

// ---------------------------------------------------------------------------
// Branch stage 4: softmax numerators/denominators. Rank-1 trick: scatter only
// sum(ex * x[src]) per (dst, head); expand by W at finalize.
// ---------------------------------------------------------------------------
__global__ void k_branch_edge_sum(const float* __restrict__ x, const int* __restrict__ ei,
                                  const float* __restrict__ ea,
                                  const unsigned char* m0, const unsigned char* m1,
                                  const unsigned char* m2, const float* __restrict__ c,
                                  const unsigned* __restrict__ m_br,
                                  float* den, float* s_br, int N, int E) {
  int e = blockIdx.x * blockDim.x + threadIdx.x;
  if (e >= E) return;
  int src = ei[e], dst = ei[E + e];
  float xs = x[src], xd = x[dst];
  float a0 = ea[e * 3 + 0], a1 = ea[e * 3 + 1], a2 = ea[e * 3 + 2];
  const unsigned char* M[3] = {m0, m1, m2};
  for (int p = 0; p < 3; ++p) {
    if (!M[p][e]) continue;
    for (int h = 0; h < 4; ++h) {
      float al = xs * c[p * 4 + h] + xd * c[16 + p * 4 + h]
               + a0 * c[32 + p * 12 + 0 * 4 + h]
               + a1 * c[32 + p * 12 + 1 * 4 + h]
               + a2 * c[32 + p * 12 + 2 * 4 + h];
      al = leaky02(al);
      size_t idx = (size_t)p * N * 4 + (size_t)dst * 4 + h;
      float ex = expf(al - dec_f(m_br[idx]));
      atomicAdd(&den[idx], ex);
      atomicAdd(&s_br[idx], ex * xs);
    }
  }
}

// ---------------------------------------------------------------------------
// Branch stage 5: finalize -> hcat[N x 192] (ht | hg | hp), each +bias.
// ---------------------------------------------------------------------------
__global__ void k_branch_final(const float* __restrict__ x, const float* __restrict__ c,
                               const float* __restrict__ attr_sum, const float* __restrict__ cnt,
                               const unsigned* __restrict__ m_br, const float* __restrict__ den,
                               const float* __restrict__ s_br,
                               const float* W0, const float* b0,
                               const float* W1, const float* b1,
                               const float* W2p, const float* b2p,
                               float* __restrict__ hcat, int N) {
  int i = blockIdx.x * blockDim.x + threadIdx.x;
  if (i >= N) return;
  const float* W[3] = {W0, W1, W2p};
  const float* B[3] = {b0, b1, b2p};
  float xi = x[i];
  for (int p = 0; p < 3; ++p) {
    float la[3];
    float ct = fmaxf(cnt[(size_t)p * N + i], 1.0f);
    for (int j = 0; j < 3; ++j) la[j] = attr_sum[(size_t)p * N * 3 + (size_t)i * 3 + j] / ct;
    for (int h = 0; h < 4; ++h) {
      float ls = xi * (c[p * 4 + h] + c[16 + p * 4 + h]);
      for (int j = 0; j < 3; ++j) ls += la[j] * c[32 + p * 12 + j * 4 + h];
      ls = leaky02(ls);
      size_t idx = (size_t)p * N * 4 + (size_t)i * 4 + h;
      float exl = expf(ls - dec_f(m_br[idx]));
      float d = den[idx] + exl;
      float coef = (s_br[idx] + exl * xi) / d;
      float* o = hcat + (size_t)i * 192 + p * 64 + h * 16;
      for (int cc = 0; cc < 16; ++cc) o[cc] = coef * W[p][h * 16 + cc] + B[p][h * 16 + cc];
    }
  }
}

// ---------------------------------------------------------------------------
// Tall-skinny fp32 GEMM with V_WMMA_F32_16X16X4_F32.
// Grid.x = M/16 tiles; one wave per 16-col output tile (blockDim = 32*Ncol/16).
// A tile (16 x K) is staged in LDS once per block and shared by all waves.
// VGPR layouts per CDNA5 ISA 7.12.2:
//   A 16x4 f32: lanes 0-15 hold K=k,k+1; lanes 16-31 hold K=k+2,k+3 (rows M=lane%16)
//   B 4x16 f32: v0 = row k (lanes 0-15) / row k+2 (lanes 16-31); v1 = k+1 / k+3
//   C/D 16x16:  VGPR j, lane l -> row j + 8*(l>=16), col l%16
// ---------------------------------------------------------------------------
__global__ void k_gemm_wmma(const float* __restrict__ A, const float* __restrict__ Bm,
                            const float* __restrict__ bias, float* __restrict__ C,
                            int M, int K, int Ncol, int relu) {
  extern __shared__ float sA[];  // 16 * K floats
  int row0 = blockIdx.x * 16;
  for (int idx = threadIdx.x; idx < 16 * K; idx += blockDim.x) {
    int r = idx / K, cc = idx % K;
    int row = row0 + r; if (row >= M) row = M - 1;
    sA[idx] = A[(size_t)row * K + cc];
  }
  __syncthreads();

  int wv = threadIdx.x >> 5;
  int lane = threadIdx.x & 31;
  int half = lane >> 4;     // 0: K-pair {k,k+1}, 1: {k+2,k+3}
  int l16 = lane & 15;
  int col0 = wv * 16;

  v8f acc = {};
  for (int k = 0; k < K; k += 4) {
    int ka = k + 2 * half;
    v2f a, b;
    a.x = sA[l16 * K + ka];
    a.y = sA[l16 * K + ka + 1];
    b.x = Bm[(size_t)ka * Ncol + col0 + l16];
    b.y = Bm[(size_t)(ka + 1) * Ncol + col0 + l16];
    acc = __builtin_amdgcn_wmma_f32_16x16x4_f32(false, a, false, b, (short)0, acc,
                                                false, false);
  }
  for (int j = 0; j < 8; ++j) {
    int row = row0 + j + 8 * half;
    if (row >= M) continue;
    int col = col0 + l16;
    float v = acc[j] + (bias ? bias[col] : 0.0f);
    if (relu) v = fmaxf(v, 0.0f);
    C[(size_t)row * Ncol + col] = v;
  }
}

// ---------------------------------------------------------------------------
// conv2/conv3 node prep: as/ad reductions, self-loop score -> max init,
// zero den + numerator buffer.
// ---------------------------------------------------------------------------
__global__ void k_conv_node(const float* __restrict__ hx, const float* __restrict__ a_s,
                            const float* __restrict__ a_d, float* as_n, float* ad_n,
                            unsigned* m2, float* den2, float* num, int N) {
  int i = blockIdx.x * blockDim.x + threadIdx.x;
  if (i >= N) return;
  const float* hrow = hx + (size_t)i * 128;
  for (int h = 0; h < 4; ++h) {
    float sa = 0.f, sd = 0.f;
    for (int cc = 0; cc < 32; ++cc) {
      float v = hrow[h * 32 + cc];
      sa += v * a_s[h * 32 + cc];
      sd += v * a_d[h * 32 + cc];
    }
    as_n[(size_t)i * 4 + h] = sa;
    ad_n[(size_t)i * 4 + h] = sd;
    float lp = leaky02(sa + sd);
    m2[(size_t)i * 4 + h] = enc_f(lp);
    den2[(size_t)i * 4 + h] = 0.f;
  }
  float* nrow = num + (size_t)i * 128;
  for (int k = 0; k < 128; ++k) nrow[k] = 0.f;
}

__global__ void k_conv_edge_max(const int* __restrict__ ei, const float* __restrict__ as_n,
                                const float* __restrict__ ad_n, unsigned* m2, int E) {
  int e = blockIdx.x * blockDim.x + threadIdx.x;
  if (e >= E) return;
  int src = ei[e], dst = ei[E + e];
  for (int h = 0; h < 4; ++h) {
    float al = leaky02(as_n[(size_t)src * 4 + h] + ad_n[(size_t)dst * 4 + h]);
    atomicMax(&m2[(size_t)dst * 4 + h], enc_f(al));
  }
}

// 4 threads per edge: thread handles one head (32-wide feature scatter)
__global__ void k_conv_edge_sum(const int* __restrict__ ei, const float* __restrict__ as_n,
                                const float* __restrict__ ad_n, const unsigned* __restrict__ m2,
                                const float* __restrict__ hx, float* den2, float* num, int E) {
  long tid = (long)blockIdx.x * blockDim.x + threadIdx.x;
  long e = tid >> 2;
  int h = (int)(tid & 3);
  if (e >= E) return;
  int src = ei[e], dst = ei[E + e];
  float al = leaky02(as_n[(size_t)src * 4 + h] + ad_n[(size_t)dst * 4 + h]);
  float ex = expf(al - dec_f(m2[(size_t)dst * 4 + h]));
  atomicAdd(&den2[(size_t)dst * 4 + h], ex);
  const float* hs = hx + (size_t)src * 128 + h * 32;
  float* nd = num + (size_t)dst * 128 + h * 32;
  for (int cc = 0; cc < 32; ++cc) atomicAdd(&nd[cc], ex * hs[cc]);
}

__global__ void k_conv_final(const float* __restrict__ hx, const float* __restrict__ as_n,
                             const float* __restrict__ ad_n, const unsigned* __restrict__ m2,
                             const float* __restrict__ den2, const float* __restrict__ num,
                             const float* __restrict__ bias, float* __restrict__ out, int N) {
  int i = blockIdx.x * blockDim.x + threadIdx.x;
  if (i >= N) return;
  for (int h = 0; h < 4; ++h) {
    float lp = leaky02(as_n[(size_t)i * 4 + h] + ad_n[(size_t)i * 4 + h]);
    float exl = expf(lp - dec_f(m2[(size_t)i * 4 + h]));
    float d = den2[(size_t)i * 4 + h] + exl;
    const float* nrow = num + (size_t)i * 128 + h * 32;
    const float* hrow = hx + (size_t)i * 128 + h * 32;
    float* orow = out + (size_t)i * 128 + h * 32;
    for (int cc = 0; cc < 32; ++cc) {
      float v = (nrow[cc] + exl * hrow[cc]) / d + bias[h * 32 + cc];
      orow[cc] = fmaxf(v, 0.0f);  // reference applies relu after conv2/conv3
    }
  }
}

// ---------------------------------------------------------------------------
// Classifier tail: logits = z1 @ Wc2 + bc2 (64 -> 5), then log_softmax.
// ---------------------------------------------------------------------------
__global__ void k_classifier(const float* __restrict__ z1, const float* __restrict__ Wc2,
                             const float* __restrict__ bc2, float* __restrict__ out, int N) {
  int i = blockIdx.x * blockDim.x + threadIdx.x;
  if (i >= N) return;
  const float* z = z1 + (size_t)i * 64;
  float lg[5];
  for (int j = 0; j < 5; ++j) {
    float s = bc2[j];
    for (int k = 0; k < 64; ++k) s += z[k] * Wc2[k * 5 + j];
    lg[j] = s;
  }
  float mx = lg[0];
  for (int j = 1; j < 5; ++j) mx = fmaxf(mx, lg[j]);
  float se = 0.f;
  for (int j = 0; j < 5; ++j) se += expf(lg[j] - mx);
  float lse = mx + logf(se);
  for (int j = 0; j < 5; ++j) out[(size_t)i * 5 + j] = lg[j] - lse;
}

// ---------------------------------------------------------------------------
extern "C" void kernel_launch(void* const* d_in, const int* in_sizes, int n_in,
                              void* d_out, int out_size, void* d_ws, size_t ws_size,
                              hipStream_t stream) {
  const float* x  = (const float*)d_in[0];
  const int* ei   = (const int*)d_in[1];
  const float* ea = (const float*)d_in[2];
  const unsigned char* mt = (const unsigned char*)d_in[3];  // jax bool = 1 byte
  const unsigned char* mg = (const unsigned char*)d_in[4];
  const unsigned char* mp = (const unsigned char*)d_in[5];
  const float *Wt = (const float*)d_in[6],  *ats = (const float*)d_in[7],
              *atd = (const float*)d_in[8], *ate = (const float*)d_in[9],
              *Wte = (const float*)d_in[10], *bt = (const float*)d_in[11];
  const float *Wg = (const float*)d_in[12], *ags = (const float*)d_in[13],
              *agd = (const float*)d_in[14], *age = (const float*)d_in[15],
              *Wge = (const float*)d_in[16], *bg = (const float*)d_in[17];
  const float *Wp = (const float*)d_in[18], *aps = (const float*)d_in[19],
              *apd = (const float*)d_in[20], *ape = (const float*)d_in[21],
              *Wpe = (const float*)d_in[22], *bp = (const float*)d_in[23];
  const float *Wf = (const float*)d_in[24], *bf = (const float*)d_in[25];
  const float *W2 = (const float*)d_in[26], *a2s = (const float*)d_in[27],
              *a2d = (const float*)d_in[28], *b2 = (const float*)d_in[29];
  const float *W3 = (const float*)d_in[30], *a3s = (const float*)d_in[31],
              *a3d = (const float*)d_in[32], *b3 = (const float*)d_in[33];
  const float *Wc1 = (const float*)d_in[34], *bc1 = (const float*)d_in[35];
  const float *Wc2 = (const float*)d_in[36], *bc2 = (const float*)d_in[37];

  const int N = in_sizes[0];
  const int E = in_sizes[3];

  // ---- workspace layout (floats) ----
  float* w = (float*)d_ws;
  size_t off = 0;
  float* c_const  = w + off; off += 128;
  float* attr_sum = w + off; off += (size_t)3 * N * 3;
  float* cnt      = w + off; off += (size_t)3 * N;
  unsigned* m_br  = (unsigned*)(w + off); off += (size_t)3 * N * 4;
  float* den_br   = w + off; off += (size_t)3 * N * 4;
  float* s_br     = w + off; off += (size_t)3 * N * 4;
  float* hcat     = w + off; off += (size_t)N * 192;  // reused: conv2 out + z1
  float* hB       = w + off; off += (size_t)N * 128;  // h (fused), later conv3 out
  float* hxC      = w + off; off += (size_t)N * 128;  // transformed features
  float* numD     = w + off; off += (size_t)N * 128;  // scatter numerator
  float* as_n     = w + off; off += (size_t)N * 4;
  float* ad_n     = w + off; off += (size_t)N * 4;
  unsigned* m2    = (unsigned*)(w + off); off += (size_t)N * 4;
  float* den2     = w + off; off += (size_t)N * 4;
  float* h2 = hcat;                         // conv2 output (hcat is dead by then)
  float* z1 = hcat + (size_t)N * 128;       // classifier hidden (N*64 fits)

  const int TB = 256;
  const int gN = (N + TB - 1) / TB;
  const int gE = (E + TB - 1) / TB;
  const int gE4 = (int)(((size_t)E * 4 + TB - 1) / TB);
  const int Mtiles = (N + 15) / 16;

  // ---- branch GATs (rank-1) ----
  k_setup<<<1, 64, 0, stream>>>(c_const, Wt, ats, atd, ate, Wte,
                                Wg, ags, agd, age, Wge, Wp, aps, apd, ape, Wpe);
  hipMemsetAsync(attr_sum, 0, ((size_t)3 * N * 3 + (size_t)3 * N) * sizeof(float), stream);
  k_attr_accum<<<gE, TB, 0, stream>>>(ei, ea, mt, mg, mp, attr_sum, cnt, N, E);
  k_branch_node<<<gN, TB, 0, stream>>>(x, c_const, attr_sum, cnt, m_br, den_br, s_br, N);
  k_branch_edge_max<<<gE, TB, 0, stream>>>(x, ei, ea, mt, mg, mp, c_const, m_br, N, E);
  k_branch_edge_sum<<<gE, TB, 0, stream>>>(x, ei, ea, mt, mg, mp, c_const, m_br,
                                           den_br, s_br, N, E);
  k_branch_final<<<gN, TB, 0, stream>>>(x, c_const, attr_sum, cnt, m_br, den_br, s_br,
                                        Wt, bt, Wg, bg, Wp, bp, hcat, N);

  // ---- fusion: h = relu(hcat @ Wf + bf)  [N x 192] x [192 x 128] ----
  k_gemm_wmma<<<Mtiles, 256, 16 * 192 * sizeof(float), stream>>>(
      hcat, Wf, bf, hB, N, 192, 128, 1);

  // ---- conv2: hx = h @ W2, GAT aggregate, +b2, relu ----
  k_gemm_wmma<<<Mtiles, 256, 16 * 128 * sizeof(float), stream>>>(
      hB, W2, nullptr, hxC, N, 128, 128, 0);
  k_conv_node<<<gN, TB, 0, stream>>>(hxC, a2s, a2d, as_n, ad_n, m2, den2, numD, N);
  k_conv_edge_max<<<gE, TB, 0, stream>>>(ei, as_n, ad_n, m2, E);
  k_conv_edge_sum<<<gE4, TB, 0, stream>>>(ei, as_n, ad_n, m2, hxC, den2, numD, E);
  k_conv_final<<<gN, TB, 0, stream>>>(hxC, as_n, ad_n, m2, den2, numD, b2, h2, N);

  // ---- conv3: same with W3 (input h2, output hB) ----
  k_gemm_wmma<<<Mtiles, 256, 16 * 128 * sizeof(float), stream>>>(
      h2, W3, nullptr, hxC, N, 128, 128, 0);
  k_conv_node<<<gN, TB, 0, stream>>>(hxC, a3s, a3d, as_n, ad_n, m2, den2, numD, N);
  k_conv_edge_max<<<gE, TB, 0, stream>>>(ei, as_n, ad_n, m2, E);
  k_conv_edge_sum<<<gE4, TB, 0, stream>>>(ei, as_n, ad_n, m2, hxC, den2, numD, E);
  k_conv_final<<<gN, TB, 0, stream>>>(hxC, as_n, ad_n, m2, den2, numD, b3, hB, N);

  // ---- classifier: z1 = relu(h3 @ Wc1 + bc1); out = log_softmax(z1 @ Wc2 + bc2) ----
  k_gemm_wmma<<<Mtiles, 128, 16 * 128 * sizeof(float), stream>>>(
      hB, Wc1, bc1, z1, N, 128, 64, 1);
  k_classifier<<<gN, TB, 0, stream>>>(z1, Wc2, bc2, (float*)d_out, N);
}